// GCN5_13065290514766
// MI455X (gfx1250) — compile-verified
//
#include <hip/hip_runtime.h>
#include <math.h>

typedef float v2f __attribute__((ext_vector_type(2)));
typedef float v8f __attribute__((ext_vector_type(8)));

#define ASTRIDE_PAD 4   // A rows stride fi+4 -> conflict-free ds reads
#define WSTRIDE_PAD 8   // W rows stride fo+8 -> lane-halves on disjoint banks

// ---------------- utility kernels ----------------

__global__ void k_zero_i32(int* __restrict__ p, int n) {
    int i = blockIdx.x * blockDim.x + threadIdx.x;
    if (i < n) p[i] = 0;
}

__global__ void k_zero_f32(float* __restrict__ p, int n) {
    int i = blockIdx.x * blockDim.x + threadIdx.x;
    if (i < n) p[i] = 0.0f;
}

// deg[d] += 1 for every edge target (self loop handled as +1 in k_dinv)
__global__ void k_deg(const int* __restrict__ dst, int* __restrict__ deg, int E) {
    int e = blockIdx.x * blockDim.x + threadIdx.x;
    if (e < E) atomicAdd(&deg[dst[e]], 1);
}

__global__ void k_dinv(const int* __restrict__ deg, float* __restrict__ dinv, int n) {
    int i = blockIdx.x * blockDim.x + threadIdx.x;
    if (i < n) dinv[i] = rsqrtf((float)(deg[i] + 1));  // +1 = self loop
}

// ---------------- prefix-scan for CSR rowptr ----------------

__global__ void k_blocksum(const int* __restrict__ deg, int* __restrict__ bsum, int n) {
    __shared__ int s[256];
    int i = blockIdx.x * 256 + threadIdx.x;
    s[threadIdx.x] = (i < n) ? deg[i] : 0;
    __syncthreads();
    for (int o = 128; o > 0; o >>= 1) {
        if (threadIdx.x < o) s[threadIdx.x] += s[threadIdx.x + o];
        __syncthreads();
    }
    if (threadIdx.x == 0) bsum[blockIdx.x] = s[0];
}

__global__ void k_scantop(int* __restrict__ bsum, int nb) {
    if (blockIdx.x == 0 && threadIdx.x == 0) {
        int acc = 0;
        for (int b = 0; b < nb; ++b) { int v = bsum[b]; bsum[b] = acc; acc += v; }
    }
}

__global__ void k_scanfinal(const int* __restrict__ deg, const int* __restrict__ bsum,
                            int* __restrict__ rowptr, int n, int E) {
    __shared__ int s[256];
    int i = blockIdx.x * 256 + threadIdx.x;
    int v = (i < n) ? deg[i] : 0;
    s[threadIdx.x] = v;
    __syncthreads();
    for (int o = 1; o < 256; o <<= 1) {
        int t = (threadIdx.x >= o) ? s[threadIdx.x - o] : 0;
        __syncthreads();
        s[threadIdx.x] += t;
        __syncthreads();
    }
    if (i < n) rowptr[i] = bsum[blockIdx.x] + s[threadIdx.x] - v;  // exclusive
    if (blockIdx.x == 0 && threadIdx.x == 0) rowptr[n] = E;
}

// ---------------- CSR fill: bucket incoming edges per target node ------------
__global__ void k_fill(const int* __restrict__ src, const int* __restrict__ dst,
                       const float* __restrict__ dinv, const int* __restrict__ rowptr,
                       int* __restrict__ cursor, int* __restrict__ col,
                       float* __restrict__ wgt, int E) {
    int e = blockIdx.x * blockDim.x + threadIdx.x;
    if (e >= E) return;
    int s = src[e], d = dst[e];
    int pos = rowptr[d] + atomicAdd(&cursor[d], 1);
    col[pos] = s;
    wgt[pos] = dinv[s] * dinv[d];
}

// ---------------- WMMA GEMM: T[n,fo] = H[n,fi] @ W[fi,fo] ----------------
// One wave = one 16x16 output tile; wave w handles column tile w.
// blockDim.x = 32*(fo/16). W and the 16 A rows staged in LDS (bank-padded).
__global__ void k_gemm_wmma(const float* __restrict__ H, const float* __restrict__ W,
                            float* __restrict__ T, int n, int fi, int fo) {
    __shared__ float Ws[128 * (128 + WSTRIDE_PAD)];  // 69632 B max
    __shared__ float As[16 * (128 + ASTRIDE_PAD)];   //  8448 B max

    const int wstr = fo + WSTRIDE_PAD;
    const int astr = fi + ASTRIDE_PAD;
    const int rowBase = blockIdx.x << 4;

    // cooperative stage of W (row-major, padded stride)
    for (int idx = threadIdx.x; idx < fi * fo; idx += blockDim.x) {
        int k = idx / fo, c = idx % fo;
        Ws[k * wstr + c] = W[idx];
    }
    // cooperative stage of 16 A rows (clamped so loads stay in range)
    for (int idx = threadIdx.x; idx < 16 * fi; idx += blockDim.x) {
        int r = idx / fi, c = idx % fi;
        int rr = rowBase + r; if (rr > n - 1) rr = n - 1;
        As[r * astr + c] = H[(size_t)rr * fi + c];
    }
    __syncthreads();

    const int lane    = threadIdx.x & 31;
    const int wave    = threadIdx.x >> 5;
    const int colBase = wave << 4;
    const int half    = lane >> 4;   // 0: K={0,1} | 1: K={2,3}
    const int l       = lane & 15;

    const float* arow = As + l * astr;
    v8f acc = {};
    for (int k = 0; k < fi; k += 4) {
        const int ka = k + (half << 1);
        v2f a, b;
        a[0] = arow[ka];
        a[1] = arow[ka + 1];
        b[0] = Ws[ka * wstr + colBase + l];
        b[1] = Ws[(ka + 1) * wstr + colBase + l];
        acc = __builtin_amdgcn_wmma_f32_16x16x4_f32(
            /*neg_a=*/false, a, /*neg_b=*/false, b,
            /*c_mod=*/(short)0, acc, /*reuse_a=*/false, /*reuse_b=*/false);
    }

    #pragma unroll
    for (int j = 0; j < 8; ++j) {
        const int m = rowBase + j + (half << 3);
        if (m < n) T[(size_t)m * fo + colBase + l] = acc[j];
    }
}

// ---------------- fused aggregation: gather-reduce + self loop + bias + ReLU
// One wave per (node, 32-feature chunk). Edge metadata loaded coalesced in
// batches of 32 and broadcast lane-to-lane; feature gathers are 128B/wave L2 hits.
__global__ void k_aggregate(const float* __restrict__ T, const int* __restrict__ rowptr,
                            const int* __restrict__ col, const float* __restrict__ wgt,
                            const float* __restrict__ dinv, const float* __restrict__ bias,
                            float* __restrict__ Hn, int n, int fo) {
    const int gtid   = blockIdx.x * blockDim.x + threadIdx.x;
    const int lane   = gtid & 31;
    const int wid    = gtid >> 5;
    const int chunks = fo >> 5;
    const int i      = wid / chunks;
    const int ch     = wid % chunks;
    if (i >= n) return;
    const int f = (ch << 5) + lane;

    const float di = dinv[i];
    float acc = di * di * T[(size_t)i * fo + f];   // self loop

    const int e0 = rowptr[i], e1 = rowptr[i + 1];  // wave-uniform
    for (int eb = e0; eb < e1; eb += 32) {
        const int ee = eb + lane;
        int   c = 0;
        float w = 0.0f;
        if (ee < e1) { c = col[ee]; w = wgt[ee]; }
        int cnt = e1 - eb; if (cnt > 32) cnt = 32;
        for (int j = 0; j < cnt; ++j) {
            const int   cj = __shfl(c, j, 32);
            const float wj = __shfl(w, j, 32);
            acc += wj * T[(size_t)cj * fo + f];
        }
    }
    Hn[(size_t)i * fo + f] = fmaxf(acc + bias[f], 0.0f);
}

// ---------------- pooling: sums[g,:] += H[i,:], cnt[g] += 1 ----------------
__global__ void k_pool(const float* __restrict__ H, const int* __restrict__ batch,
                       float* __restrict__ sums, float* __restrict__ cnts, int n) {
    long long tid = (long long)blockIdx.x * blockDim.x + threadIdx.x;
    int i = (int)(tid >> 5);
    int f = (int)(tid & 31);
    if (i >= n) return;
    int g = batch[i];
    atomicAdd(&sums[g * 32 + f], H[(size_t)i * 32 + f]);
    if (f == 0) atomicAdd(&cnts[g], 1.0f);
}

// ---------------- final FC ----------------
__global__ void k_fc(const float* __restrict__ sums, const float* __restrict__ cnts,
                     const float* __restrict__ Wfc, const float* __restrict__ bfc,
                     float* __restrict__ out, int nGraphs, int nClasses) {
    int tid = blockIdx.x * blockDim.x + threadIdx.x;
    if (tid >= nGraphs * nClasses) return;
    int g = tid / nClasses;
    int c = tid % nClasses;
    float inv = 1.0f / fmaxf(cnts[g], 1.0f);
    float acc = bfc[c];
    for (int k = 0; k < 32; ++k)
        acc += (sums[g * 32 + k] * inv) * Wfc[k * nClasses + c];
    out[tid] = acc;
}

// ---------------- host launch ----------------

static inline unsigned cdiv_ll(long long a, long long b) {
    return (unsigned)((a + b - 1) / b);
}

extern "C" void kernel_launch(void* const* d_in, const int* in_sizes, int n_in,
                              void* d_out, int out_size, void* d_ws, size_t ws_size,
                              hipStream_t stream) {
    const float* x     = (const float*)d_in[0];
    const int*   ei    = (const int*)d_in[1];
    const int*   batch = (const int*)d_in[2];
    const float* Wl[5] = {(const float*)d_in[3], (const float*)d_in[5], (const float*)d_in[7],
                          (const float*)d_in[9], (const float*)d_in[11]};
    const float* bl[5] = {(const float*)d_in[4], (const float*)d_in[6], (const float*)d_in[8],
                          (const float*)d_in[10], (const float*)d_in[12]};
    const float* Wfc = (const float*)d_in[13];
    const float* bfc = (const float*)d_in[14];

    const int n = in_sizes[0] / 128;         // 100000 nodes
    const int E = in_sizes[1] / 2;           // 1.6M edges
    const int nGraphs  = 64;
    const int nClasses = in_sizes[13] / 32;  // 10
    const int nb = (n + 255) / 256;

    const int* src = ei;
    const int* dst = ei + E;

    // ---- workspace layout (256B aligned) ----
    char* ws = (char*)d_ws;
    size_t off = 0;
    auto take = [&](size_t bytes) {
        char* p = ws + off;
        off += (bytes + 255) & ~(size_t)255;
        return p;
    };
    int*   deg    = (int*)take((size_t)n * 4);
    int*   cursor = (int*)take((size_t)n * 4);
    int*   rowptr = (int*)take((size_t)(n + 1) * 4);
    int*   bsum   = (int*)take((size_t)nb * 4);
    float* dinv   = (float*)take((size_t)n * 4);
    int*   csrcol = (int*)take((size_t)E * 4);
    float* csrwgt = (float*)take((size_t)E * 4);
    float* sums   = (float*)take((size_t)nGraphs * 32 * 4);
    float* cnts   = (float*)take((size_t)nGraphs * 4);
    float* B1     = (float*)take((size_t)n * 128 * 4);  // T
    float* B2     = (float*)take((size_t)n * 128 * 4);  // H ping
    float* B3     = (float*)take((size_t)n * 128 * 4);  // H pong
    (void)ws_size;

    // ---- graph preprocessing: degrees, norms, CSR (once per launch) ----
    k_zero_i32<<<cdiv_ll(n, 256), 256, 0, stream>>>(deg, n);
    k_zero_i32<<<cdiv_ll(n, 256), 256, 0, stream>>>(cursor, n);
    k_deg<<<cdiv_ll(E, 256), 256, 0, stream>>>(dst, deg, E);
    k_dinv<<<cdiv_ll(n, 256), 256, 0, stream>>>(deg, dinv, n);
    k_blocksum<<<nb, 256, 0, stream>>>(deg, bsum, n);
    k_scantop<<<1, 1, 0, stream>>>(bsum, nb);
    k_scanfinal<<<nb, 256, 0, stream>>>(deg, bsum, rowptr, n, E);
    k_fill<<<cdiv_ll(E, 256), 256, 0, stream>>>(src, dst, dinv, rowptr, cursor,
                                                csrcol, csrwgt, E);

    // ---- 5 GCN layers: WMMA GEMM + fused gather-aggregate ----
    const int fis[5] = {128, 128, 128, 128, 64};
    const int fos[5] = {128, 128, 128, 64, 32};
    float* pp[2] = {B2, B3};
    const float* Hcur = x;

    for (int L = 0; L < 5; ++L) {
        const int fi = fis[L], fo = fos[L];
        float* T  = B1;
        float* Hn = pp[L & 1];

        k_gemm_wmma<<<cdiv_ll(n, 16), 32 * (fo / 16), 0, stream>>>(Hcur, Wl[L], T, n, fi, fo);
        k_aggregate<<<cdiv_ll((long long)n * fo, 256), 256, 0, stream>>>(
            T, rowptr, csrcol, csrwgt, dinv, bl[L], Hn, n, fo);

        Hcur = Hn;
    }

    // ---- global mean pool + FC ----
    k_zero_f32<<<cdiv_ll(nGraphs * 32, 256), 256, 0, stream>>>(sums, nGraphs * 32);
    k_zero_f32<<<1, 64, 0, stream>>>(cnts, nGraphs);
    k_pool<<<cdiv_ll((long long)n * 32, 256), 256, 0, stream>>>(Hcur, batch, sums, cnts, n);
    k_fc<<<cdiv_ll(nGraphs * nClasses, 128), 128, 0, stream>>>(sums, cnts, Wfc, bfc,
                                                               (float*)d_out, nGraphs, nClasses);
}